// TFNEigenNet_64905545777697
// MI455X (gfx1250) — compile-verified
//
#include <hip/hip_runtime.h>
#include <math.h>

// ---------- CDNA5 WMMA fragment types (wave32) ----------
typedef float v2f __attribute__((ext_vector_type(2)));
typedef float v4f __attribute__((ext_vector_type(4)));
typedef float v8f __attribute__((ext_vector_type(8)));
// pointer element type expected by the async Global->LDS builtins in this toolchain
typedef int asy4i __attribute__((__vector_size__(4 * sizeof(int))));

#define N_ATOMC 17
#define NPAIR   289        // 17*17
#define MPAD    304        // pairs padded to 19*16
#define RH      32         // radial hidden width
#define LD      2048       // layer width
#define CAT0K   4096
#define CAT1K   6144
#define KSPLIT  4          // K-split for self-interaction GEMMs (deterministic partials)
#define EPSF    1e-8f
#define LOG2F_  0.69314718055994530942f

// Async Global->LDS path (gfx1250): direct memory->LDS DMA tracked by ASYNCcnt,
// enabling double-buffered weight streaming that overlaps HBM latency with WMMA.
#if defined(__has_builtin)
#  if __has_builtin(__builtin_amdgcn_global_load_async_to_lds_b128) && \
      __has_builtin(__builtin_amdgcn_s_wait_asynccnt)
#    define USE_ASYNC_LDS 1
#  endif
#endif
#ifndef USE_ASYNC_LDS
#  define USE_ASYNC_LDS 0
#endif

// ---------- workspace layout (float offsets) ----------
// total ~4.43M floats (~17.7 MB)
constexpr size_t OFF_RADIAL = 0;        // MPAD*RH          = 9728
constexpr size_t OFF_UNIT   = 10240;    // NPAIR*4 (ux,uy,uz,mask) = 1156
constexpr size_t OFF_L1SCR  = 11520;    // 17 + 51
constexpr size_t OFF_L0     = 12288;    // 17*LD            = 34816
constexpr size_t OFF_L1     = 47360;    // 17*LD*3          = 104448
constexpr size_t OFF_CAT0   = 152064;   // 32*CAT0K         = 131072
constexpr size_t OFF_CAT1   = 283136;   // 64*CAT1K         = 393216
constexpr size_t OFF_Y0P    = 676352;   // KSPLIT*32*LD     = 262144
constexpr size_t OFF_Y1P    = 938496;   // KSPLIT*64*LD     = 524288
constexpr size_t OFF_R      = 1462784;  // 5*NPAIR*LD       = 2959360

__device__ __forceinline__ float sp_(float x) {           // softplus
  return (x > 20.f) ? x : log1pf(expf(x));
}
__device__ __forceinline__ float ssp_(float x) {          // shifted softplus
  return sp_(x) - LOG2F_;
}

// ---------- zero-fill (pads for WMMA A matrices) ----------
__global__ void k_zero(float* __restrict__ p, int n) {
  int i = blockIdx.x * blockDim.x + threadIdx.x;
  if (i < n) p[i] = 0.f;
}

// ---------- geometry + radial MLP: radial[289,32], unit4[289]{ux,uy,uz,mask} ----------
__global__ void k_geom(const float* __restrict__ xin,
                       const float* __restrict__ w1, const float* __restrict__ b1,
                       const float* __restrict__ w2, const float* __restrict__ b2,
                       const float* __restrict__ w3, const float* __restrict__ b3,
                       float* __restrict__ radial, float* __restrict__ unit4) {
  int p = blockIdx.x * blockDim.x + threadIdx.x;
  if (p >= NPAIR) return;
  int a = p / N_ATOMC, b = p % N_ATOMC;
  float ax = 0.f, ay = 0.f, az = 0.f, bx = 0.f, by = 0.f, bz = 0.f;
  if (a > 0) { ax = xin[(a-1)*3+0]; ay = xin[(a-1)*3+1]; az = xin[(a-1)*3+2]; }
  if (b > 0) { bx = xin[(b-1)*3+0]; by = xin[(b-1)*3+1]; bz = xin[(b-1)*3+2]; }
  float rx = ax - bx, ry = ay - by, rz = az - bz;
  float d2 = rx*rx + ry*ry + rz*rz;
  float dij = sqrtf(fmaxf(d2, EPSF));
  float inv = 1.f / dij;
  unit4[p*4+0] = rx * inv;
  unit4[p*4+1] = ry * inv;
  unit4[p*4+2] = rz * inv;
  unit4[p*4+3] = (d2 >= EPSF) ? 1.f : 0.f;

  float h1[RH], h2[RH];
#pragma unroll
  for (int j = 0; j < RH; ++j) h1[j] = sp_(dij * w1[j] + b1[j]);
#pragma unroll
  for (int j = 0; j < RH; ++j) {
    float s = b2[j];
#pragma unroll
    for (int k = 0; k < RH; ++k) s += h1[k] * w2[k*RH + j];
    h2[j] = sp_(s);
  }
#pragma unroll
  for (int j = 0; j < RH; ++j) {
    float s = b3[j];
#pragma unroll
    for (int k = 0; k < RH; ++k) s += h2[k] * w3[k*RH + j];
    radial[(size_t)p*RH + j] = sp_(s);
  }
}

// ---------- five filter GEMMs: R[m][289,2048] = radial[304,32] @ W[32,2048] + b (mask m=1,3,4) ----------
struct FilterArgs { const float* w[5]; const float* b[5]; };

__global__ void k_rfilt(const float* __restrict__ radial, const float* __restrict__ unit4,
                        FilterArgs fa, float* __restrict__ R) {
  const int m    = blockIdx.z;
  const int Mb   = blockIdx.y * 16;
  const int Nb   = blockIdx.x * 16;
  const int lane = threadIdx.x;          // one wave32 per block
  const int half = lane >> 4;            // 0|1
  const int l16  = lane & 15;
  const float* __restrict__ W = fa.w[m];
  const float* __restrict__ B = fa.b[m];

  v8f acc = {0.f,0.f,0.f,0.f,0.f,0.f,0.f,0.f};
#pragma unroll
  for (int t = 0; t < 8; ++t) {          // K = 32 = 8 x (16x16x4)
    const int k = t*4 + half*2;
    v2f af, bf;
    af.x = radial[(size_t)(Mb + l16)*RH + k];
    af.y = radial[(size_t)(Mb + l16)*RH + k + 1];
    bf.x = W[(size_t)k       * LD + Nb + l16];
    bf.y = W[(size_t)(k + 1) * LD + Nb + l16];
    acc = __builtin_amdgcn_wmma_f32_16x16x4_f32(false, af, false, bf, (short)0, acc, false, false);
  }
  const float bias = B[Nb + l16];
  const bool useMask = (m == 1) || (m == 3) || (m == 4);
#pragma unroll
  for (int v = 0; v < 8; ++v) {
    const int r = Mb + v + half*8;       // C/D layout: row = vgpr + 8*(lane>=16)
    if (r < NPAIR) {
      float val = acc[v] + bias;
      if (useMask) val *= unit4[r*4 + 3];
      R[((size_t)m*NPAIR + r)*LD + Nb + l16] = val;
    }
  }
}

// ---------- per-pair Clebsch-Gordan contractions over neighbor b ----------
__global__ void k_pair(const float* __restrict__ R, const float* __restrict__ unit4,
                       const float* __restrict__ l0, const float* __restrict__ l1,
                       float* __restrict__ cat0, float* __restrict__ cat1) {
  int idx = blockIdx.x * blockDim.x + threadIdx.x;
  if (idx >= N_ATOMC * LD) return;
  const int a = idx / LD, n = idx % LD;
  float o0a = 0.f, o0b = 0.f;
  float o1a[3] = {0,0,0}, o1b[3] = {0,0,0}, o1c[3] = {0,0,0};
  for (int b = 0; b < N_ATOMC; ++b) {
    const int p = a * N_ATOMC + b;
    const size_t pn = (size_t)p * LD + n;
    const float r0 = R[pn];
    const float r1 = R[(size_t)1*NPAIR*LD + pn];
    const float r2 = R[(size_t)2*NPAIR*LD + pn];
    const float r3 = R[(size_t)3*NPAIR*LD + pn];
    const float r4 = R[(size_t)4*NPAIR*LD + pn];
    const float ux = unit4[p*4+0], uy = unit4[p*4+1], uz = unit4[p*4+2];
    const float t0 = l0[(size_t)b*LD + n];
    const size_t lb = ((size_t)b*LD + n)*3;
    const float tx = l1[lb], ty = l1[lb+1], tz = l1[lb+2];
    o0a += r0 * t0;                                   // 0x0 -> 0
    const float s = r1 * t0;                          // 0x1 -> 1
    o1a[0] += ux*s; o1a[1] += uy*s; o1a[2] += uz*s;
    o1b[0] += r2*tx; o1b[1] += r2*ty; o1b[2] += r2*tz; // 1x0 -> 1
    o0b += r3 * (ux*tx + uy*ty + uz*tz);              // 1x1 -> 0
    o1c[0] += r4 * (uy*tz - uz*ty);                   // 1x1 -> 1 (Levi-Civita)
    o1c[1] += r4 * (uz*tx - ux*tz);
    o1c[2] += r4 * (ux*ty - uy*tx);
  }
  cat0[(size_t)a*CAT0K + n]      = o0a;
  cat0[(size_t)a*CAT0K + LD + n] = o0b;
#pragma unroll
  for (int i = 0; i < 3; ++i) {
    const size_t row = (size_t)(a*3 + i);
    cat1[row*CAT1K +          n] = o1a[i];
    cat1[row*CAT1K +   LD   + n] = o1b[i];
    cat1[row*CAT1K + 2*LD   + n] = o1c[i];
  }
}

// ---------- self-interaction GEMM: y = A[MT*16, KDIM] x W^T, K-split partials ----------
// W is [2048 rows(g) x KDIM cols(f)]; B[k][n] = W[n*KDIM + k].
// Weight stream double-buffered through LDS via async Global->LDS DMA when available.
template <int MT, int KDIM>
__global__ void k_si(const float* __restrict__ A, const float* __restrict__ Wt,
                     float* __restrict__ ypart) {
  __shared__ float lb[2][16][68];        // 2 x (16 n-rows x 64 k), pad 68 (no bank conflicts)
  const int Nb   = blockIdx.x * 16;
  const int ks   = blockIdx.y;
  const int lane = threadIdx.x;          // one wave32 per block
  const int half = lane >> 4;
  const int l16  = lane & 15;
  const int KSEG = KDIM / KSPLIT;
  const int kbeg = ks * KSEG;

  // stage one 4KB weight chunk (16 rows x 64 floats) into LDS buffer `buf`
  auto stage = [&](int kcS, int bufS) {
#pragma unroll
    for (int q = 0; q < 8; ++q) {
      const int idx  = q*32 + lane;
      const int row  = idx >> 4;
      const int quad = idx & 15;
      const float* src = &Wt[(size_t)(Nb + row)*KDIM + kcS + quad*4];
#if USE_ASYNC_LDS
      __builtin_amdgcn_global_load_async_to_lds_b128(
          (asy4i*)src, (asy4i*)&lb[bufS][row][quad*4],
          /*offset=*/0, /*cpol=*/0);
#else
      const v4f v = *(const v4f*)src;
      lb[bufS][row][quad*4+0] = v.x; lb[bufS][row][quad*4+1] = v.y;
      lb[bufS][row][quad*4+2] = v.z; lb[bufS][row][quad*4+3] = v.w;
#endif
    }
  };

  v8f acc[MT];
  const v8f z = {0.f,0.f,0.f,0.f,0.f,0.f,0.f,0.f};
#pragma unroll
  for (int m = 0; m < MT; ++m) acc[m] = z;

  int cur = 0;
  stage(kbeg, 0);                        // prologue: chunk 0 in flight
  for (int kc = kbeg; kc < kbeg + KSEG; kc += 64) {
    const bool hasNext = (kc + 64 < kbeg + KSEG);
    if (hasNext) stage(kc + 64, cur ^ 1);   // keep next chunk streaming
#if USE_ASYNC_LDS
    // async loads complete in order: <=8 outstanding means current chunk landed
    if (hasNext) __builtin_amdgcn_s_wait_asynccnt(8);
    else         __builtin_amdgcn_s_wait_asynccnt(0);
#else
    __syncthreads();
    if (hasNext)                          // hint next weight chunk toward the caches
      __builtin_prefetch(&Wt[(size_t)(Nb + l16)*KDIM + kc + 64 + half*32], 0, 1);
#endif
#pragma unroll
    for (int t = 0; t < 16; ++t) {        // 16 x (16x16x4) per chunk
      const int kl = t*4 + half*2;
      v2f bf;
      bf.x = lb[cur][l16][kl];
      bf.y = lb[cur][l16][kl + 1];
      const int kg = kc + kl;
#pragma unroll
      for (int m = 0; m < MT; ++m) {
        v2f af;
        af.x = A[(size_t)(m*16 + l16)*KDIM + kg];
        af.y = A[(size_t)(m*16 + l16)*KDIM + kg + 1];
        acc[m] = __builtin_amdgcn_wmma_f32_16x16x4_f32(false, af, false, bf, (short)0, acc[m], false, false);
      }
    }
    cur ^= 1;
  }
#pragma unroll
  for (int m = 0; m < MT; ++m)
#pragma unroll
    for (int v = 0; v < 8; ++v) {
      const int r = m*16 + v + half*8;
      ypart[((size_t)ks*(MT*16) + r)*LD + Nb + l16] = acc[m][v];
    }
}

// ---------- layer 1 (1-channel embed): per-atom reductions ----------
__global__ void k_l1a(const float* __restrict__ radial, const float* __restrict__ unit4,
                      const float* __restrict__ f0w, const float* __restrict__ f0b,
                      const float* __restrict__ f1w, const float* __restrict__ f1b,
                      const float* __restrict__ embedw, float* __restrict__ scr) {
  int a = threadIdx.x;
  if (a >= N_ATOMC) return;
  const float e = embedw[0];
  float s0 = 0.f, s1x = 0.f, s1y = 0.f, s1z = 0.f;
  for (int b = 0; b < N_ATOMC; ++b) {
    const int p = a * N_ATOMC + b;
    float d0 = f0b[0], d1 = f1b[0];
#pragma unroll
    for (int k = 0; k < RH; ++k) {
      const float rv = radial[(size_t)p*RH + k];
      d0 += rv * f0w[k];
      d1 += rv * f1w[k];
    }
    s0 += d0 * e;
    const float s = d1 * unit4[p*4+3] * e;
    s1x += unit4[p*4+0] * s;
    s1y += unit4[p*4+1] * s;
    s1z += unit4[p*4+2] * s;
  }
  scr[a] = s0;
  scr[N_ATOMC + a*3 + 0] = s1x;
  scr[N_ATOMC + a*3 + 1] = s1y;
  scr[N_ATOMC + a*3 + 2] = s1z;
}

__global__ void k_l1b(const float* __restrict__ scr,
                      const float* __restrict__ si0w, const float* __restrict__ si0b,
                      const float* __restrict__ si1w, const float* __restrict__ nlb,
                      float* __restrict__ l0, float* __restrict__ l1) {
  int idx = blockIdx.x * blockDim.x + threadIdx.x;
  if (idx >= N_ATOMC * LD) return;
  const int a = idx / LD, g = idx % LD;
  l0[(size_t)a*LD + g] = ssp_(scr[a] * si0w[g] + si0b[g]);
  const float vx = scr[N_ATOMC + a*3 + 0] * si1w[g];
  const float vy = scr[N_ATOMC + a*3 + 1] * si1w[g];
  const float vz = scr[N_ATOMC + a*3 + 2] * si1w[g];
  const float nrm = sqrtf(fmaxf(vx*vx + vy*vy + vz*vz, EPSF));
  const float fac = ssp_(nrm + nlb[g]) / nrm;
  const size_t o = ((size_t)a*LD + g)*3;
  l1[o+0] = vx*fac; l1[o+1] = vy*fac; l1[o+2] = vz*fac;
}

// ---------- reduce K-split partials, add bias, equivariant nonlinearity ----------
__global__ void k_post(const float* __restrict__ y0p, const float* __restrict__ y1p,
                       const float* __restrict__ si0b, const float* __restrict__ nlb,
                       float* __restrict__ l0, float* __restrict__ l1) {
  int idx = blockIdx.x * blockDim.x + threadIdx.x;
  if (idx >= N_ATOMC * LD) return;
  const int a = idx / LD, g = idx % LD;
  float y0 = si0b[g];
#pragma unroll
  for (int ks = 0; ks < KSPLIT; ++ks) y0 += y0p[((size_t)ks*32 + a)*LD + g];
  l0[(size_t)a*LD + g] = ssp_(y0);
  float v[3];
#pragma unroll
  for (int i = 0; i < 3; ++i) {
    float s = 0.f;
#pragma unroll
    for (int ks = 0; ks < KSPLIT; ++ks) s += y1p[((size_t)ks*64 + a*3 + i)*LD + g];
    v[i] = s;
  }
  const float nrm = sqrtf(fmaxf(v[0]*v[0] + v[1]*v[1] + v[2]*v[2], EPSF));
  const float fac = ssp_(nrm + nlb[g]) / nrm;
  const size_t o = ((size_t)a*LD + g)*3;
  l1[o+0] = v[0]*fac; l1[o+1] = v[1]*fac; l1[o+2] = v[2]*fac;
}

// ---------- final readout ----------
__global__ void k_final(const float* __restrict__ l0, const float* __restrict__ l1,
                        const float* __restrict__ xin, const float* __restrict__ finalw,
                        const float* __restrict__ elecw, float* __restrict__ out) {
  int c = blockIdx.x * blockDim.x + threadIdx.x;
  if (c >= LD) return;
  float s = 0.f;
  for (int a = 0; a < N_ATOMC; ++a) {
    s += l0[(size_t)a*LD + c];
    const size_t o = ((size_t)a*LD + c)*3;
    s += l1[o] + l1[o+1] + l1[o+2];
  }
  const float fw = finalw[0] + finalw[1] + finalw[2] + finalw[3];
  float ew = 0.f;
  for (int i = 0; i < N_ATOMC; ++i) ew += elecw[i];
  float d = 1.f;
  for (int t = 0; t < 48; ++t) {
    const float v = xin[t];
    d *= (sqrtf(200.f - v*v) - 10.f) * 0.1f;   // lim=10, D_avg=0
  }
  out[c] = s * fw * ew * d;
}

// ---------- orchestration ----------
extern "C" void kernel_launch(void* const* d_in, const int* in_sizes, int n_in,
                              void* d_out, int out_size, void* d_ws, size_t ws_size,
                              hipStream_t stream) {
  (void)in_sizes; (void)n_in; (void)out_size; (void)ws_size;
  auto in = [&](int i) { return (const float*)d_in[i]; };
  float* ws = (float*)d_ws;

  float* radial = ws + OFF_RADIAL;
  float* unit4  = ws + OFF_UNIT;
  float* l1scr  = ws + OFF_L1SCR;
  float* l0buf  = ws + OFF_L0;
  float* l1buf  = ws + OFF_L1;
  float* cat0   = ws + OFF_CAT0;
  float* cat1   = ws + OFF_CAT1;
  float* y0p    = ws + OFF_Y0P;
  float* y1p    = ws + OFF_Y1P;
  float* Rbuf   = ws + OFF_R;

  // zero WMMA A-matrix pad rows (valid rows are rewritten every call)
  k_zero<<<(MPAD*RH + 255)/256, 256, 0, stream>>>(radial, MPAD*RH);
  k_zero<<<(32*CAT0K + 255)/256, 256, 0, stream>>>(cat0, 32*CAT0K);
  k_zero<<<(64*CAT1K + 255)/256, 256, 0, stream>>>(cat1, 64*CAT1K);

  // geometry + shared radial MLP  (radial params: b,w pairs at indices 40..45)
  k_geom<<<(NPAIR + 127)/128, 128, 0, stream>>>(in(0),
      in(41), in(40), in(43), in(42), in(45), in(44), radial, unit4);

  // layer 1 (embed width 1): indices 4..11
  k_l1a<<<1, 32, 0, stream>>>(radial, unit4, in(5), in(4), in(7), in(6), in(2), l1scr);
  k_l1b<<<(N_ATOMC*LD + 255)/256, 256, 0, stream>>>(l1scr, in(10), in(9), in(11), in(8),
                                                    l0buf, l1buf);

  // heavy layers 2 and 3 (14 leaves each: base 12 and 26)
  const int bases[2] = {12, 26};
  for (int L = 0; L < 2; ++L) {
    const int b = bases[L];
    FilterArgs fa;
    for (int m = 0; m < 5; ++m) { fa.b[m] = in(b + 2*m); fa.w[m] = in(b + 2*m + 1); }
    k_rfilt<<<dim3(LD/16, MPAD/16, 5), 32, 0, stream>>>(radial, unit4, fa, Rbuf);
    k_pair<<<(N_ATOMC*LD + 255)/256, 256, 0, stream>>>(Rbuf, unit4, l0buf, l1buf, cat0, cat1);
    k_si<2, CAT0K><<<dim3(LD/16, KSPLIT), 32, 0, stream>>>(cat0, in(b + 12), y0p);
    k_si<4, CAT1K><<<dim3(LD/16, KSPLIT), 32, 0, stream>>>(cat1, in(b + 13), y1p);
    k_post<<<(N_ATOMC*LD + 255)/256, 256, 0, stream>>>(y0p, y1p, in(b + 11), in(b + 10),
                                                       l0buf, l1buf);
  }

  // final readout -> out[2048]
  k_final<<<(LD + 255)/256, 256, 0, stream>>>(l0buf, l1buf, in(0), in(3), in(1),
                                              (float*)d_out);
}